// CLS_80530636800126
// MI455X (gfx1250) — compile-verified
//
#include <hip/hip_runtime.h>

#define D_IN  128
#define D_HID 64   // [W1 | W2] fused output width
#define D_OUT 32

typedef float v2f __attribute__((ext_vector_type(2)));
typedef float v8f __attribute__((ext_vector_type(8)));

// ---------------------------------------------------------------------------
// h[N,64] = x[N,128] @ [W1 | W2]  using V_WMMA_F32_16X16X4_F32 (wave32).
// One wave computes a 16-row x 64-col stripe (4 accumulator tiles).
// LDS holds W K-pair-interleaved: ldsW[(k>>1)*128 + col*2 + (k&1)] = W[k][col]
// so each B fragment (rows kk,kk+1 @ one column) is a single ds_load_b64.
// ---------------------------------------------------------------------------
__global__ __launch_bounds__(128) void k_gemm(const float* __restrict__ x,
                                              const float* __restrict__ W1,
                                              const float* __restrict__ W2,
                                              float* __restrict__ h, int N)
{
    __shared__ float ldsW[D_IN * D_HID];   // 32 KB
    const int tid = threadIdx.x;
    for (int idx = tid; idx < D_IN * D_HID; idx += 128) {
        const int p = idx >> 7;            // K pair
        const int c = (idx & 127) >> 1;    // column 0..63
        const int q = idx & 1;             // parity within pair
        const int k = 2 * p + q;
        ldsW[idx] = (c < 32) ? W1[k * 32 + c] : W2[k * 32 + (c - 32)];
    }
    __syncthreads();

    const int lane = tid & 31;
    const int half = lane >> 4;   // 0: holds K+0/K+1, 1: holds K+2/K+3 (ISA A layout)
    const int l15  = lane & 15;
    const int row0 = blockIdx.x * 64 + (tid >> 5) * 16;

    int arow = row0 + l15;
    if (arow >= N) arow = N - 1;          // clamp (keeps EXEC all-ones for WMMA)
    const float* xrow = x + (size_t)arow * D_IN + half * 2;

    v8f acc[4] = {};
    #pragma unroll 4
    for (int k0 = 0; k0 < D_IN; k0 += 4) {
        const v2f a = *(const v2f*)(xrow + k0);               // A frag: 2 K-values/lane
        // B frag base: K-pair index (k0/2 + half), column l15
        const int kb = (k0 >> 1) * 128 + half * 128 + l15 * 2;
        #pragma unroll
        for (int t = 0; t < 4; ++t) {
            const v2f b = *(const v2f*)&ldsW[kb + t * 32];    // one ds_load_b64
            acc[t] = __builtin_amdgcn_wmma_f32_16x16x4_f32(
                false, a, false, b, (short)0, acc[t], false, false);
        }
    }

    // C/D layout: VGPR j -> rows (row0 + j) [lanes 0-15] / (row0 + 8 + j) [lanes 16-31]
    float* hb = h + (size_t)(row0 + half * 8) * D_HID + l15;
    if (row0 + 16 <= N) {
        // fast path: all 16 rows valid -> 32 plain b32 stores, immediate offsets
        #pragma unroll
        for (int t = 0; t < 4; ++t)
            #pragma unroll
            for (int j = 0; j < 8; ++j)
                hb[j * D_HID + t * 16] = acc[t][j];
    } else {
        #pragma unroll
        for (int t = 0; t < 4; ++t)
            #pragma unroll
            for (int j = 0; j < 8; ++j)
                if (row0 + half * 8 + j < N) hb[j * D_HID + t * 16] = acc[t][j];
    }
}

// deg[i] = 1 (self loop)
__global__ void k_deg_init(float* __restrict__ deg, int N)
{
    const int i = blockIdx.x * blockDim.x + threadIdx.x;
    if (i < N) deg[i] = 1.0f;
}

// deg[dst] += 1 per edge
__global__ void k_deg_edges(const int* __restrict__ dst, float* __restrict__ deg, int E)
{
    const int e = blockIdx.x * blockDim.x + threadIdx.x;
    if (e < E) unsafeAtomicAdd(&deg[dst[e]], 1.0f);
}

// deg -> rsqrt(deg) in place (deg >= 1 always, so no zero guard needed)
__global__ void k_dinv(float* __restrict__ deg, int N)
{
    const int i = blockIdx.x * blockDim.x + threadIdx.x;
    if (i < N) deg[i] = rsqrtf(deg[i]);
}

// acc init with self-loop contribution: acc[i,:] = h[i,:] * dinv[i]^2
__global__ void k_self(const float* __restrict__ h, const float* __restrict__ dinv,
                       float* __restrict__ acc, int N)
{
    const int idx = blockIdx.x * blockDim.x + threadIdx.x;
    if (idx < N * D_HID) {
        const float di = dinv[idx >> 6];
        acc[idx] = h[idx] * di * di;
    }
}

// Edge scatter: acc[dst,:] += h[src,:] * dinv[src]*dinv[dst]; 32 lanes/edge, 2 floats/lane.
__global__ __launch_bounds__(256) void k_scatter(const float* __restrict__ h,
                                                 const float* __restrict__ dinv,
                                                 const int* __restrict__ src,
                                                 const int* __restrict__ dst,
                                                 float* __restrict__ acc, int E)
{
    const long long t = (long long)blockIdx.x * 256 + threadIdx.x;
    const long long e = t >> 5;
    const int lane = (int)(t & 31);
    if (e >= E) return;
    const int s = src[e];
    const int d = dst[e];
    const float w = dinv[s] * dinv[d];
    const v2f hv = *(const v2f*)(h + (size_t)s * D_HID + lane * 2);
    float* out = acc + (size_t)d * D_HID + lane * 2;
    unsafeAtomicAdd(out,     hv.x * w);
    unsafeAtomicAdd(out + 1, hv.y * w);
}

// Bias add, split into x1/x2, log_softmax(x1+x2) with wave32 shuffle reductions.
// One wave per node (32 lanes == 32 features).
__global__ __launch_bounds__(256) void k_final(const float* __restrict__ acc,
                                               const float* __restrict__ b1,
                                               const float* __restrict__ b2,
                                               float* __restrict__ out, int N)
{
    const int lane = threadIdx.x & 31;
    const int n = blockIdx.x * 8 + (threadIdx.x >> 5);
    if (n >= N) return;
    const float x1 = acc[(size_t)n * D_HID + lane]      + b1[lane];
    const float x2 = acc[(size_t)n * D_HID + 32 + lane] + b2[lane];
    const float s = x1 + x2;

    float m = s;
    #pragma unroll
    for (int off = 16; off > 0; off >>= 1) m = fmaxf(m, __shfl_xor(m, off, 32));
    float sum = expf(s - m);
    #pragma unroll
    for (int off = 16; off > 0; off >>= 1) sum += __shfl_xor(sum, off, 32);
    const float lse = m + logf(sum);

    out[(size_t)n * D_OUT + lane] = s - lse;                             // x3
    out[(size_t)N * D_OUT + (size_t)n * D_OUT + lane] = x1;              // x1
    out[(size_t)2 * N * D_OUT + (size_t)n * D_OUT + lane] = x2;          // x2
}

extern "C" void kernel_launch(void* const* d_in, const int* in_sizes, int n_in,
                              void* d_out, int out_size, void* d_ws, size_t ws_size,
                              hipStream_t stream)
{
    const float* x  = (const float*)d_in[0];
    const float* W1 = (const float*)d_in[1];
    const float* b1 = (const float*)d_in[2];
    const float* W2 = (const float*)d_in[3];
    const float* b2 = (const float*)d_in[4];
    const int*   ei = (const int*)d_in[5];

    const int N = in_sizes[0] / D_IN;
    const int E = in_sizes[5] / 2;
    const int* src = ei;
    const int* dst = ei + E;

    // Workspace layout: h[N*64] | acc[N*64] | deg/dinv[N]  (~51.6 MB)
    float* h    = (float*)d_ws;
    float* accb = h + (size_t)N * D_HID;
    float* deg  = accb + (size_t)N * D_HID;

    k_gemm<<<(N + 63) / 64, 128, 0, stream>>>(x, W1, W2, h, N);
    k_deg_init<<<(N + 255) / 256, 256, 0, stream>>>(deg, N);
    k_deg_edges<<<(E + 255) / 256, 256, 0, stream>>>(dst, deg, E);
    k_dinv<<<(N + 255) / 256, 256, 0, stream>>>(deg, N);
    k_self<<<(int)(((size_t)N * D_HID + 255) / 256), 256, 0, stream>>>(h, deg, accb, N);
    const long long tot = (long long)E * 32;
    k_scatter<<<(int)((tot + 255) / 256), 256, 0, stream>>>(h, deg, src, dst, accb, E);
    k_final<<<(N + 7) / 8, 256, 0, stream>>>(accb, b1, b2, (float*)d_out, N);
}